// DenseCapRoIHeads_57827439673366
// MI455X (gfx1250) — compile-verified
//
#include <hip/hip_runtime.h>
#include <math.h>

#define BB   16
#define NN   16000
#define GG   128
#define NPG  (NN + GG)            // 16128 = 63 * 256
#define GPB  8                    // gt rows per block in kernel B
#define IMG_WF 800.0f
#define IMG_HF 800.0f
#define CLIP_V 4.135166556742356f // log(1000/16)

// ---- CDNA5 async global->LDS staging (guarded; falls back to sync copy) ----
#if defined(__has_builtin)
#  if __has_builtin(__builtin_amdgcn_global_load_async_to_lds_b128)
#    define HAS_ASYNC_LDS 1
#  endif
#endif
#ifndef HAS_ASYNC_LDS
#  define HAS_ASYNC_LDS 0
#endif

typedef int v4i __attribute__((vector_size(16)));           // matches builtin param
typedef __attribute__((address_space(1))) v4i* gv4p;        // global AS pointer
typedef __attribute__((address_space(3))) v4i* lv4p;        // LDS AS pointer

__device__ __forceinline__ void stage16(const float4* g, float4* l) {
#if HAS_ASYNC_LDS
  __builtin_amdgcn_global_load_async_to_lds_b128((gv4p)(const void*)g,
                                                 (lv4p)(void*)l,
                                                 /*offset=*/0, /*cpol=*/0);
#else
  *l = *g;
#endif
}

__device__ __forceinline__ void stage_wait() {
#if HAS_ASYNC_LDS
#  if __has_builtin(__builtin_amdgcn_s_wait_asynccnt)
  __builtin_amdgcn_s_wait_asynccnt(0);
#  else
  asm volatile("s_wait_asynccnt 0" ::: "memory");
#  endif
#endif
}

// ---------------- Kernel A: decode+clip pred_boxes, labels ------------------
// grid = (NPG/256, B), block = 256 (8 wave32)
__global__ __launch_bounds__(256) void kernelA(
    const float* __restrict__ proposals,   // (B,N,4)
    const float* __restrict__ gt_boxes,    // (B,G,4)
    const float* __restrict__ box_reg,     // (B,N+G,4)
    float* __restrict__ pred,              // (B,N+G,4)
    float* __restrict__ labels) {          // (B,N+G)
  __shared__ float4 s_gt[GG];
  const int t = threadIdx.x;
  const int b = blockIdx.y;
  const int j = blockIdx.x * 256 + t;

  const float4* gtb = (const float4*)(gt_boxes + (size_t)b * GG * 4);
  if (t < GG) stage16(gtb + t, &s_gt[t]);
  stage_wait();
  __syncthreads();
  if (j >= NPG) return;

  // column box from prop_all = concat(proposals, gt_boxes)
  float4 p;
  if (j < NN) p = ((const float4*)(proposals + (size_t)b * NN * 4))[j];
  else        p = s_gt[j - NN];
  const float pw  = p.z - p.x, ph  = p.w - p.y;
  const float pcx = p.x + 0.5f * pw, pcy = p.y + 0.5f * ph;
  const float parea = pw * ph;

  // decode + clip (fused: reuses the column box)
  float4 r = ((const float4*)(box_reg + (size_t)b * NPG * 4))[j];
  float dx = r.x * 0.1f;
  float dy = r.y * 0.1f;
  float dw = fminf(r.z * 0.2f, CLIP_V);
  float dh = fminf(r.w * 0.2f, CLIP_V);
  float ncx = dx * pw + pcx, ncy = dy * ph + pcy;
  float nw = __expf(dw) * pw, nh = __expf(dh) * ph;
  float4 o;
  o.x = fminf(fmaxf(ncx - 0.5f * nw, 0.0f), IMG_WF);
  o.y = fminf(fmaxf(ncy - 0.5f * nh, 0.0f), IMG_HF);
  o.z = fminf(fmaxf(ncx + 0.5f * nw, 0.0f), IMG_WF);
  o.w = fminf(fmaxf(ncy + 0.5f * nh, 0.0f), IMG_HF);
  ((float4*)(pred + (size_t)b * NPG * 4))[j] = o;

  // max IoU over 128 gt (from LDS)
  float mv = 0.0f;
#pragma unroll 4
  for (int g = 0; g < GG; ++g) {
    float4 q = s_gt[g];
    float garea = (q.z - q.x) * (q.w - q.y);
    float ix0 = fmaxf(p.x, q.x), iy0 = fmaxf(p.y, q.y);
    float ix1 = fminf(p.z, q.z), iy1 = fminf(p.w, q.w);
    float iw = fmaxf(ix1 - ix0, 0.0f), ih = fmaxf(iy1 - iy0, 0.0f);
    float inter = iw * ih;
    float iou = __fdividef(inter, parea + garea - inter);
    mv = fmaxf(mv, iou);
  }
  labels[(size_t)b * NPG + j] = (mv >= 0.5f) ? 1.0f : 0.0f;
}

// -------- Kernel B: per-gt argmax over N proposals, encode, scores ----------
// grid = (G/GPB, B), block = 256; proposals streamed through LDS chunks
__global__ __launch_bounds__(256) void kernelB(
    const float* __restrict__ proposals,   // (B,N,4)
    const float* __restrict__ gt_boxes,    // (B,G,4)
    float* __restrict__ regT,              // (B,G,4)
    float* __restrict__ scores) {          // (B,G)
  __shared__ float4 s_prop[256];
  __shared__ float  s_rv[8];
  __shared__ int    s_ri[8];
  const int t  = threadIdx.x;
  const int b  = blockIdx.y;
  const int g0 = blockIdx.x * GPB;
  const float4* pb  = (const float4*)(proposals + (size_t)b * NN * 4);
  const float4* gtb = (const float4*)(gt_boxes + (size_t)b * GG * 4);

  float gx0[GPB], gy0[GPB], gx1[GPB], gy1[GPB], ga[GPB];
#pragma unroll
  for (int k = 0; k < GPB; ++k) {          // uniform address -> scalar loads
    float4 q = gtb[g0 + k];
    gx0[k] = q.x; gy0[k] = q.y; gx1[k] = q.z; gy1[k] = q.w;
    ga[k] = (q.z - q.x) * (q.w - q.y);
  }
  float bv[GPB]; int bi[GPB];
#pragma unroll
  for (int k = 0; k < GPB; ++k) { bv[k] = -1.0f; bi[k] = 0; }

  const int nchunks = (NN + 255) / 256;    // 63
  for (int c = 0; c < nchunks; ++c) {
    const int j = c * 256 + t;
    __syncthreads();                       // protect s_prop before overwrite
    if (j < NN) stage16(pb + j, &s_prop[t]);
    stage_wait();
    __syncthreads();
    if ((c + 1) * 256 + t < NN)            // global_prefetch_b8 for next chunk
      __builtin_prefetch((const void*)(pb + (c + 1) * 256 + t), 0, 1);
    if (j < NN) {
      float4 p = s_prop[t];
      float parea = (p.z - p.x) * (p.w - p.y);
#pragma unroll
      for (int k = 0; k < GPB; ++k) {
        float ix0 = fmaxf(p.x, gx0[k]), iy0 = fmaxf(p.y, gy0[k]);
        float ix1 = fminf(p.z, gx1[k]), iy1 = fminf(p.w, gy1[k]);
        float iw = fmaxf(ix1 - ix0, 0.0f), ih = fmaxf(iy1 - iy0, 0.0f);
        float inter = iw * ih;
        float iou = __fdividef(inter, parea + ga[k] - inter);
        if (iou > bv[k]) { bv[k] = iou; bi[k] = j; }  // ascending j => first hit
      }
    }
  }

  const int lane = t & 31, wid = t >> 5;
  for (int k = 0; k < GPB; ++k) {
    float v = bv[k]; int idx = bi[k];
#pragma unroll
    for (int off = 16; off > 0; off >>= 1) {  // wave32 shuffle reduce
      float ov = __shfl_down(v, off);
      int   oi = __shfl_down(idx, off);
      if (ov > v || (ov == v && oi < idx)) { v = ov; idx = oi; }
    }
    if (lane == 0) { s_rv[wid] = v; s_ri[wid] = idx; }
    __syncthreads();
    if (t == 0) {
      float fv = s_rv[0]; int fi = s_ri[0];
#pragma unroll
      for (int w = 1; w < 8; ++w) {
        float ov = s_rv[w]; int oi = s_ri[w];
        if (ov > fv || (ov == fv && oi < fi)) { fv = ov; fi = oi; }
      }
      const int g = g0 + k;
      float4 mp = pb[fi];
      float pw = mp.z - mp.x, ph = mp.w - mp.y;
      float pcx = mp.x + 0.5f * pw, pcy = mp.y + 0.5f * ph;
      float gw = gx1[k] - gx0[k], gh = gy1[k] - gy0[k];
      float gcx = gx0[k] + 0.5f * gw, gcy = gy0[k] + 0.5f * gh;
      float4 e;
      e.x = 10.0f * (gcx - pcx) / pw;
      e.y = 10.0f * (gcy - pcy) / ph;
      e.z = 5.0f * __logf(gw / pw);
      e.w = 5.0f * __logf(gh / ph);
      ((float4*)regT)[(size_t)b * GG + g] = e;
      scores[(size_t)b * GG + g] = fv;
    }
    __syncthreads();
  }
}

extern "C" void kernel_launch(void* const* d_in, const int* in_sizes, int n_in,
                              void* d_out, int out_size, void* d_ws, size_t ws_size,
                              hipStream_t stream) {
  (void)in_sizes; (void)n_in; (void)out_size; (void)d_ws; (void)ws_size;
  const float* proposals = (const float*)d_in[0];  // (B,N,4)
  const float* gt_boxes  = (const float*)d_in[1];  // (B,G,4)
  const float* box_reg   = (const float*)d_in[2];  // (B,N+G,4)

  float* out    = (float*)d_out;
  float* pred   = out;                                  // B*NPG*4
  float* regT   = pred   + (size_t)BB * NPG * 4;        // B*G*4
  float* scores = regT   + (size_t)BB * GG * 4;         // B*G
  float* labels = scores + (size_t)BB * GG;             // B*NPG

  dim3 gridA(NPG / 256, BB);
  kernelA<<<gridA, 256, 0, stream>>>(proposals, gt_boxes, box_reg, pred, labels);

  dim3 gridB(GG / GPB, BB);
  kernelB<<<gridB, 256, 0, stream>>>(proposals, gt_boxes, regT, scores);
}